// ModelNew_3556232921835
// MI455X (gfx1250) — compile-verified
//
#include <hip/hip_runtime.h>
#include <math.h>

// Problem sizes (fixed by the reference)
#define M_ROWS 16384
#define K_DIM  4096
#define N_DIM  4096

typedef __attribute__((ext_vector_type(2))) float v2f;
typedef __attribute__((ext_vector_type(4))) float v4f;
typedef __attribute__((ext_vector_type(8))) float v8f;

// -------- workspace layout (floats) --------
#define WS_PARTIAL 0                    // 8 * 4096 partial column sums of W
#define WS_WBAR    (8 * K_DIM)          // 4096: column mean of W
#define WS_ADDEND  (8 * K_DIM + K_DIM)  // 1: log(N) - mean(subtract)

// -------- fused-kernel LDS layout (floats) --------
#define PITCH      4100                 // 4096 + 4 pad: bank-conflict-free WMMA A fetch
#define XS_OFF     0                    // 16 rows of x
#define WB_OFF     (16 * PITCH)         // wbar (4096)
#define PD_OFF     (WB_OFF + K_DIM)     // 8 waves * 16 partial dots
#define GL_OFF     (PD_OFF + 128)       // 16 gelu values
#define SMEM_FLOATS (GL_OFF + 16)       // 69840 floats = 279360 B  (< 320 KB)

// ============================================================
// Kernel 1: partial column sums of W (N split into 8 slices)
// grid (16, 8) x 256 threads; fully coalesced; no atomics.
// ============================================================
__global__ __launch_bounds__(256) void wcolsum_partial_kernel(
    const float* __restrict__ w, float* __restrict__ ws) {
  const int k  = blockIdx.x * 256 + threadIdx.x;
  const int nb = blockIdx.y;
  const float* col = w + k;
  float acc = 0.f;
  const int n0 = nb * 512;
  #pragma unroll 8
  for (int n = n0; n < n0 + 512; ++n) acc += col[(size_t)n * K_DIM];
  ws[WS_PARTIAL + nb * K_DIM + k] = acc;
}

// ============================================================
// Kernel 2: blocks 0..15 reduce partials -> wbar;
//           block 16 reduces subtract -> addend = log(N) - mean(subtract)
// ============================================================
__global__ __launch_bounds__(256) void prep_finalize_kernel(
    const float* __restrict__ sub, float* __restrict__ ws) {
  if (blockIdx.x < 16) {
    const int k = blockIdx.x * 256 + threadIdx.x;
    float s = 0.f;
    #pragma unroll
    for (int nb = 0; nb < 8; ++nb) s += ws[WS_PARTIAL + nb * K_DIM + k];
    ws[WS_WBAR + k] = s * (1.0f / (float)N_DIM);
  } else {
    __shared__ float sh[256];
    const int tid = threadIdx.x;
    float s = 0.f;
    #pragma unroll
    for (int j = 0; j < 16; ++j) s += sub[tid + j * 256];
    sh[tid] = s;
    __syncthreads();
    for (int off = 128; off > 0; off >>= 1) {
      if (tid < off) sh[tid] += sh[tid + off];
      __syncthreads();
    }
    if (tid == 0)
      ws[WS_ADDEND] = 8.31776616671934f /* log(4096) */
                      - sh[0] * (1.0f / (float)N_DIM);
  }
}

// ============================================================
// Kernel 3 (fused, streaming): one block = 16 rows resident in LDS.
//   stage 1: stage 16 full x rows (+ wbar) into LDS       (x read once)
//   stage 2: per-wave 512-wide K slice of 16 row-dots via
//            V_WMMA_F32_16X16X4_F32 (wbar broadcast in B columns);
//            wave offset folded into base pointers so the loop has a
//            compile-time-constant trip count (scalar branch, EXEC all-1s)
//   stage 3: cross-wave reduce, mean + addend, tanh-surrogate GELU
//   stage 4: out = gelu[m] + x[m,:] streamed from LDS      (out write once)
// ============================================================
__global__ __launch_bounds__(256) void fused_rowgelu_kernel(
    const float* __restrict__ x, const float* __restrict__ ws,
    float* __restrict__ out) {
  extern __shared__ float smem[];
  const int tid  = threadIdx.x;
  const int wave = tid >> 5;
  const int lane = tid & 31;
  const int row0 = blockIdx.x << 4;

  // ---- stage 1: cooperative staging (float4, coalesced) ----
  const v4f* __restrict__ x4  = (const v4f*)x;
  const v4f* __restrict__ wb4 = (const v4f*)(ws + WS_WBAR);
  #pragma unroll 4
  for (int i = 0; i < 64; ++i) {          // 16 rows * 1024 float4
    const int idx4 = tid + (i << 8);
    const int m  = idx4 >> 10;
    const int n4 = idx4 & 1023;
    v4f v = x4[(size_t)(row0 + m) * 1024 + n4];
    *(v4f*)&smem[XS_OFF + m * PITCH + (n4 << 2)] = v;
  }
  #pragma unroll
  for (int i = 0; i < 4; ++i) {           // wbar: 1024 float4
    const int idx4 = tid + (i << 8);
    *(v4f*)&smem[WB_OFF + (idx4 << 2)] = wb4[idx4];
  }
  __syncthreads();

  // ---- stage 2: WMMA fp32 16x16x4 row-dot over this wave's K slice ----
  // A layout: lane half h -> K base 2h, VGPR v -> K=2h+v, row = lane&15.
  // B layout: same (h,v)->K mapping; all 16 N columns get wbar (broadcast).
  const int h2   = (lane >> 4) << 1;      // 0 or 2
  const int r    = lane & 15;
  const int koff = wave << 9;             // 512-wide slice per wave
  const float* xr  = &smem[XS_OFF + r * PITCH + h2 + koff];
  const float* wbr = &smem[WB_OFF + h2 + koff];
  v8f acc0 = {0.f, 0.f, 0.f, 0.f, 0.f, 0.f, 0.f, 0.f};
  v8f acc1 = acc0;
  #pragma unroll 4
  for (int i = 0; i < 64; ++i) {          // constant trip count: uniform branch
    const int k0 = i << 3;
    v2f a0 = *(const v2f*)(xr + k0);
    v2f b0 = *(const v2f*)(wbr + k0);
    v2f a1 = *(const v2f*)(xr + k0 + 4);
    v2f b1 = *(const v2f*)(wbr + k0 + 4);
    acc0 = __builtin_amdgcn_wmma_f32_16x16x4_f32(false, a0, false, b0,
                                                 (short)0, acc0, false, false);
    acc1 = __builtin_amdgcn_wmma_f32_16x16x4_f32(false, a1, false, b1,
                                                 (short)0, acc1, false, false);
  }
  v8f acc = acc0 + acc1;

  // D layout: lane L, VGPR v -> row m = 8*(L>>4)+v (all N columns identical)
  if (lane == 0) {
    #pragma unroll
    for (int v = 0; v < 8; ++v) smem[PD_OFF + wave * 16 + v] = acc[v];
  } else if (lane == 16) {
    #pragma unroll
    for (int v = 0; v < 8; ++v) smem[PD_OFF + wave * 16 + 8 + v] = acc[v];
  }
  __syncthreads();

  // ---- stage 3: reduce across waves + fast-tanh GELU ----
  if (tid < 16) {
    float s = 0.f;
    #pragma unroll
    for (int w2 = 0; w2 < 8; ++w2) s += smem[PD_OFF + w2 * 16 + tid];
    const float c  = s * (1.0f / (float)N_DIM) + ws[WS_ADDEND];
    const float t  = 0.7978845608028654f * (c + 0.044715f * c * c * c);
    const float th = t / (fabsf(t) + 1.0f);
    smem[GL_OFF + tid] = 0.5f * c * (1.0f + th);
  }
  __syncthreads();

  // ---- stage 4: out = gelu[m] + x[m,:] (x from LDS, float4 stores) ----
  v4f* __restrict__ out4 = (v4f*)out;
  #pragma unroll 4
  for (int i = 0; i < 64; ++i) {
    const int idx4 = tid + (i << 8);
    const int m  = idx4 >> 10;
    const int n4 = idx4 & 1023;
    const float g = smem[GL_OFF + m];
    v4f xv = *(const v4f*)&smem[XS_OFF + m * PITCH + (n4 << 2)];
    out4[(size_t)(row0 + m) * 1024 + n4] = xv + g;
  }
}

// ============================================================
extern "C" void kernel_launch(void* const* d_in, const int* in_sizes, int n_in,
                              void* d_out, int out_size, void* d_ws, size_t ws_size,
                              hipStream_t stream) {
  (void)in_sizes; (void)n_in; (void)out_size; (void)ws_size;
  const float* x   = (const float*)d_in[0];
  const float* w   = (const float*)d_in[1];
  const float* sub = (const float*)d_in[2];
  float* out = (float*)d_out;
  float* ws  = (float*)d_ws;

  static_assert(SMEM_FLOATS * 4 <= 320 * 1024, "LDS budget");

  wcolsum_partial_kernel<<<dim3(16, 8), 256, 0, stream>>>(w, ws);
  prep_finalize_kernel<<<17, 256, 0, stream>>>(sub, ws);

  // allow > 64 KB dynamic LDS for the fused kernel (idempotent, host-side)
  (void)hipFuncSetAttribute(reinterpret_cast<const void*>(fused_rowgelu_kernel),
                            hipFuncAttributeMaxDynamicSharedMemorySize,
                            SMEM_FLOATS * 4);

  fused_rowgelu_kernel<<<M_ROWS / 16, 256, SMEM_FLOATS * 4, stream>>>(x, ws, out);
}